// RFLAHead_69312182222902
// MI455X (gfx1250) — compile-verified
//
#include <hip/hip_runtime.h>
#include <hip/hip_bf16.h>
#include <stdint.h>

#ifndef __has_builtin
#define __has_builtin(x) 0
#endif

#if __has_builtin(__builtin_amdgcn_global_load_async_to_lds_b128)
#define HAVE_ASYNC_LDS 1
#else
#define HAVE_ASYNC_LDS 0
#endif

__device__ __forceinline__ void wait_async0() {
#if __has_builtin(__builtin_amdgcn_s_wait_asynccnt)
    __builtin_amdgcn_s_wait_asynccnt(0);
#else
    asm volatile("s_wait_asynccnt 0" ::: "memory");
#endif
}

typedef __attribute__((ext_vector_type(2))) float v2f;
typedef __attribute__((ext_vector_type(8))) float v8f;
typedef int v4i __attribute__((vector_size(16)));         // matches builtin's int4
typedef __attribute__((address_space(1))) v4i gas_v4i;    // global ("__device__") AS
typedef __attribute__((address_space(3))) v4i las_v4i;    // LDS ("__shared__") AS

// ---------------------------------------------------------------------------
// Constants from the reference
// ---------------------------------------------------------------------------
#define BATCH     8
#define NBOX      100000
#define NCLS      80
#define NROWS     (BATCH * NBOX)      // 800000
#define PRETOPK   1000
#define AFTERTOPK 100
#define NUMDET    200
#define IMGW      1024.0f

// ===========================================================================
// Kernel A: streaming row-max over scores (256 MB, the bandwidth-bound pass).
// Each wave async-copies its 32 contiguous rows (32*80*4B = 10KB) to LDS,
// waits on ASYNCcnt, then each lane max-reduces its own row from LDS.
// ===========================================================================
__global__ __launch_bounds__(128) void rowmax_kernel(const float* __restrict__ scores,
                                                     float* __restrict__ maxs) {
    __shared__ float sbuf[128 * NCLS];          // 40 KB
    const int tid  = threadIdx.x;
    const int wave = tid >> 5;
    const int lane = tid & 31;
    const int row0 = blockIdx.x * 128 + wave * 32;   // 32 rows per wave

    const float* gbase = scores + (size_t)row0 * NCLS;   // 2560 contiguous floats
    float*       lbase = &sbuf[wave * 32 * NCLS];

#if HAVE_ASYNC_LDS
    #pragma unroll
    for (int i = 0; i < 20; ++i) {
        const int e = (i * 32 + lane) * 4;               // float index, 16B chunks
        __builtin_amdgcn_global_load_async_to_lds_b128(
            (gas_v4i*)(gbase + e),
            (las_v4i*)(lbase + e),
            /*imm offset*/ 0, /*cpol*/ 0);
    }
    wait_async0();
    const float* rp = &sbuf[(wave * 32 + lane) * NCLS];
#else
    // Fallback: straight global reads (no LDS staging).
    const float* rp = gbase + lane * NCLS;
#endif

    float4 m4 = *(const float4*)rp;
    float  m  = fmaxf(fmaxf(m4.x, m4.y), fmaxf(m4.z, m4.w));
    #pragma unroll
    for (int c = 4; c < NCLS; c += 4) {
        float4 v = *(const float4*)(rp + c);
        m = fmaxf(m, fmaxf(fmaxf(v.x, v.y), fmaxf(v.z, v.w)));
    }
    maxs[row0 + lane] = m;
}

// ===========================================================================
// Kernel B: per-batch top-1000 via 2-level radix select on f32 bit patterns
// (values are non-negative => bits are monotonic), then exact rank-sort of
// the 1000 survivors (value desc, index asc) to match lax.top_k ordering.
// One block per batch; the 400 KB/batch input is L2-resident.
// ===========================================================================
__global__ __launch_bounds__(1024) void topk1000_kernel(const float* __restrict__ maxs,
                                                        int* __restrict__ topk) {
    const int b   = blockIdx.x;
    const int tid = threadIdx.x;
    const float* v = maxs + (size_t)b * NBOX;

    __shared__ int   hist[1024];
    __shared__ float svals[1024];
    __shared__ int   sidx[1024];
    __shared__ int   sh[8];   // 0:t1 1:above1 2:t2 3:sel_exact 4:need_eq 5:cntA 6:cntB

    // ---- level 1 histogram (bits >> 20) ----
    hist[tid] = 0;
    __syncthreads();
    for (int i = tid; i < NBOX; i += 1024) {
        unsigned k = __float_as_uint(v[i]);
        atomicAdd(&hist[(k >> 20) & 1023], 1);
    }
    __syncthreads();
    if (tid == 0) {
        int cum = 0, t1 = 0, ab = 0;
        for (int i = 1023; i >= 0; --i) {
            int h = hist[i];
            if (cum + h >= PRETOPK) { t1 = i; ab = cum; break; }
            cum += h;
        }
        sh[0] = t1; sh[1] = ab;
    }
    __syncthreads();
    const int t1 = sh[0], above1 = sh[1];

    // ---- level 2 histogram within threshold bin (bits >> 10 & 0x3FF) ----
    hist[tid] = 0;
    __syncthreads();
    for (int i = tid; i < NBOX; i += 1024) {
        unsigned k = __float_as_uint(v[i]);
        if ((int)((k >> 20) & 1023) == t1) atomicAdd(&hist[(k >> 10) & 1023], 1);
    }
    __syncthreads();
    if (tid == 0) {
        const int target = PRETOPK - above1;
        int cum = 0, t2 = 0, ab = 0;
        for (int i = 1023; i >= 0; --i) {
            int h = hist[i];
            if (cum + h >= target) { t2 = i; ab = cum; break; }
            cum += h;
        }
        sh[2] = t2;
        sh[3] = above1 + ab;             // strictly-above count
        sh[4] = PRETOPK - (above1 + ab); // how many to take from == group
        sh[5] = 0; sh[6] = 0;
    }
    __syncthreads();
    const int t2 = sh[2], selx = sh[3], needeq = sh[4];

    // ---- compaction into LDS ----
    for (int i = tid; i < NBOX; i += 1024) {
        unsigned k  = __float_as_uint(v[i]);
        int g1 = (k >> 20) & 1023;
        int g2 = (k >> 10) & 1023;
        bool top = (g1 > t1) || (g1 == t1 && g2 > t2);
        bool eq  = (g1 == t1 && g2 == t2);
        if (top) {
            int p = atomicAdd(&sh[5], 1);
            svals[p] = v[i]; sidx[p] = i;
        } else if (eq) {
            int p = atomicAdd(&sh[6], 1);
            if (p < needeq) { svals[selx + p] = v[i]; sidx[selx + p] = i; }
        }
    }
    __syncthreads();

    // ---- exact rank sort of the 1000 survivors ----
    if (tid < PRETOPK) {
        const float vj = svals[tid];
        const int   ij = sidx[tid];
        int rank = 0;
        for (int k2 = 0; k2 < PRETOPK; ++k2) {
            float vk = svals[k2];
            int   ik = sidx[k2];
            rank += (vk > vj) || (vk == vj && ik < ij);
        }
        topk[b * 1024 + rank] = ij;
    }
}

// ===========================================================================
// Kernel C: finalize. The mask has <=200 nonzeros total, so the per-batch
// top-100 reduces to ranking <=200 candidates. Boxes for the candidates are
// computed via the affine map  box = P*M2 + D*M1  expressed as two chained
// V_WMMA_F32_16X16X4_F32 (then clipped in VALU). One block per batch.
// ===========================================================================
__global__ __launch_bounds__(256) void finalize_kernel(const float* __restrict__ pts,
                                                       const float* __restrict__ bbox,
                                                       const float* __restrict__ scores,
                                                       const long long* __restrict__ sel,
                                                       const int* __restrict__ topk,
                                                       float* __restrict__ out) {
    const int b   = blockIdx.x;
    const int tid = threadIdx.x;

    __shared__ int   sb[NUMDET], scc[NUMDET], sxx[NUMDET];
    __shared__ int   cc[NUMDET], cx[NUMDET];
    __shared__ float cs[NUMDET];
    __shared__ float cpx[NUMDET], cpy[NUMDET];
    __shared__ float cd0[NUMDET], cd1[NUMDET], cd2[NUMDET], cd3[NUMDET];
    __shared__ float cbox[NUMDET][4];
    __shared__ int   s_cnum;

    if (tid < NUMDET) {
        sb[tid]  = (int)sel[tid * 3 + 0];
        scc[tid] = (int)sel[tid * 3 + 1];
        sxx[tid] = (int)sel[tid * 3 + 2];
    }
    if (tid == 0) s_cnum = 0;
    __syncthreads();

    // Dedupe + stable compaction of candidates belonging to this batch;
    // duplicates become a multiplicity weight (the .at[].add(1.0) semantics).
    if (tid < NUMDET && sb[tid] == b) {
        bool dup = false;
        int  w   = 0;
        for (int s = 0; s < NUMDET; ++s) {
            bool same = (sb[s] == b && scc[s] == scc[tid] && sxx[s] == sxx[tid]);
            if (same) { ++w; if (s < tid) dup = true; }
        }
        if (!dup) {
            int pos = 0;
            for (int s = 0; s < tid; ++s) {
                if (sb[s] != b) continue;
                bool d2 = false;
                for (int u = 0; u < s; ++u)
                    if (sb[u] == b && scc[u] == scc[s] && sxx[u] == sxx[s]) { d2 = true; break; }
                if (!d2) ++pos;
            }
            const int c   = scc[tid];
            const int x   = sxx[tid];
            const int row = topk[b * 1024 + x];
            const size_t rbase = (size_t)b * NBOX + row;
            cc[pos] = c;
            cx[pos] = x;
            cs[pos] = scores[rbase * NCLS + c] * (float)w;
            cpx[pos] = pts[rbase * 2 + 0];
            cpy[pos] = pts[rbase * 2 + 1];
            cd0[pos] = bbox[rbase * 4 + 0];
            cd1[pos] = bbox[rbase * 4 + 1];
            cd2[pos] = bbox[rbase * 4 + 2];
            cd3[pos] = bbox[rbase * 4 + 3];
            atomicAdd(&s_cnum, 1);
        }
    }
    __syncthreads();
    const int cnum = s_cnum;

    // ---- WMMA: boxes for candidate tiles of 16 (wave 0 only, EXEC all-ones)
    if (tid < 32) {
        const int lane = tid;
        const int half = lane >> 4;        // K half for A/B, row half for D
        const int m    = lane & 15;        // A row / B column / D column index
        const int ntiles = (cnum + 15) / 16;
        for (int t = 0; t < ntiles; ++t) {
            const int j  = t * 16 + m;
            const bool ok = (j < cnum);
            v2f ad, ap, b1, b2;
            // A (16x4 f32): lanes 0-15 hold K=0,1 ; lanes 16-31 hold K=2,3
            ad.x = ok ? (half ? cd2[j] : cd0[j]) : 0.f;
            ad.y = ok ? (half ? cd3[j] : cd1[j]) : 0.f;
            ap.x = (ok && !half) ? cpx[j] : 0.f;   // points occupy K=0,1 only
            ap.y = (ok && !half) ? cpy[j] : 0.f;
            // B (4x16): M1 = diag(-1,-1,1,1) ; M2[0]=[1,0,1,0], M2[1]=[0,1,0,1]
            b1.x = half ? ((m == 2) ? 1.f : 0.f) : ((m == 0) ? -1.f : 0.f);
            b1.y = half ? ((m == 3) ? 1.f : 0.f) : ((m == 1) ? -1.f : 0.f);
            b2.x = half ? 0.f : ((m == 0 || m == 2) ? 1.f : 0.f);
            b2.y = half ? 0.f : ((m == 1 || m == 3) ? 1.f : 0.f);

            v8f acc = {};
            acc = __builtin_amdgcn_wmma_f32_16x16x4_f32(false, ad, false, b1,
                                                        (short)0, acc, false, false);
            acc = __builtin_amdgcn_wmma_f32_16x16x4_f32(false, ap, false, b2,
                                                        (short)0, acc, false, false);

            // D 16x16 f32: vgpr v holds row v (lanes<16) / v+8 (lanes>=16), col = lane%16
            if (m < 4) {
                #pragma unroll
                for (int vv = 0; vv < 8; ++vv) {
                    const int jj = t * 16 + vv + (half ? 8 : 0);
                    if (jj < cnum) {
                        float val = acc[vv];
                        val = fminf(fmaxf(val, 0.f), IMGW);   // clip (W == H)
                        cbox[jj][m] = val;
                    }
                }
            }
        }
    }
    __syncthreads();

    // ---- rank candidates (score desc, flat index asc) and emit ----
    float* dets   = out;                         // [8][100][5]
    float* labels = out + BATCH * AFTERTOPK * 5; // [8][100]
    if (tid < cnum) {
        const int   flat = cc[tid] * PRETOPK + cx[tid];
        const float vj   = cs[tid];
        int rank = 0;
        for (int k2 = 0; k2 < cnum; ++k2) {
            const int   fk = cc[k2] * PRETOPK + cx[k2];
            const float vk = cs[k2];
            rank += (vk > vj) || (vk == vj && fk < flat);
        }
        if (rank < AFTERTOPK) {
            float* o = dets + ((size_t)b * AFTERTOPK + rank) * 5;
            o[0] = cbox[tid][0]; o[1] = cbox[tid][1];
            o[2] = cbox[tid][2]; o[3] = cbox[tid][3];
            o[4] = vj;
            labels[b * AFTERTOPK + rank] = (float)cc[tid];
        }
    }
    // zero-fill remaining slots (mask zeros => box 0, score 0, label idx/1000 == 0)
    if (tid < AFTERTOPK && tid >= cnum) {
        float* o = dets + ((size_t)b * AFTERTOPK + tid) * 5;
        o[0] = 0.f; o[1] = 0.f; o[2] = 0.f; o[3] = 0.f; o[4] = 0.f;
        labels[b * AFTERTOPK + tid] = 0.f;
    }
}

// ===========================================================================
// Launch
// ===========================================================================
extern "C" void kernel_launch(void* const* d_in, const int* in_sizes, int n_in,
                              void* d_out, int out_size, void* d_ws, size_t ws_size,
                              hipStream_t stream) {
    const float*     pts    = (const float*)d_in[0];
    const float*     bbox   = (const float*)d_in[1];
    const float*     scores = (const float*)d_in[2];
    const long long* sel    = (const long long*)d_in[3];

    float* maxs = (float*)d_ws;                                        // 800000 f32
    int*   topk = (int*)((char*)d_ws + (size_t)NROWS * sizeof(float)); // 8*1024 int

    rowmax_kernel<<<NROWS / 128, 128, 0, stream>>>(scores, maxs);
    topk1000_kernel<<<BATCH, 1024, 0, stream>>>(maxs, topk);
    finalize_kernel<<<BATCH, 256, 0, stream>>>(pts, bbox, scores, sel, topk,
                                               (float*)d_out);
}